// MoELayer_89455578841617
// MI455X (gfx1250) — compile-verified
//
#include <hip/hip_runtime.h>
#include <hip/hip_bf16.h>

typedef __attribute__((ext_vector_type(2))) float v2f;
typedef __attribute__((ext_vector_type(4))) float v4f;
typedef __attribute__((ext_vector_type(8))) float v8f;

// Only meaningful to check on the device pass; host (x86) never has amdgcn builtins.
#if defined(__HIP_DEVICE_COMPILE__)
#if !__has_builtin(__builtin_amdgcn_wmma_f32_16x16x4_f32)
#error "device pass: missing __builtin_amdgcn_wmma_f32_16x16x4_f32"
#endif
#endif

// Problem constants (from reference)
constexpr int kD = 1024;              // feature dim
constexpr int kE = 64;                // num experts
constexpr int kTokens = 8 * 4096;     // 32768 tokens
// Tiling
constexpr int kThreads = 256;         // 8 waves
constexpr int kTokPerWG = 128;        // 16 tokens per wave
constexpr int kKC = 64;               // K-chunk staged in LDS
// LDS strides (floats), chosen for bank-conflict-free WMMA fragment reads
constexpr int kAStride = 68;          // 272B rows: 16B aligned; frag banks 4t..4t+3 (perfect)
constexpr int kBStride = 80;          // 320B rows: half-wave offset = 32 banks
constexpr int kLStride = 65;          // per-token logit rows, conflict-free scans
constexpr int kAFloats = kTokPerWG * kAStride;        // 8704
constexpr int kBFloats = kKC * kBStride;              // 5120
constexpr int kStageFloats = kAFloats + kBFloats;     // 13824 (logits[128*65]=8320 aliases this)
constexpr int kSmemFloats = kStageFloats + kTokPerWG * 2;  // + gates -> 56320 bytes

__global__ __launch_bounds__(kThreads)
void moe_router_dispatch(const float* __restrict__ x,
                         const float* __restrict__ w,
                         float* __restrict__ out,
                         int* __restrict__ idx_out)
{
    __shared__ float smem[kSmemFloats];
    float* As     = smem;                 // [128][68] staged token chunk
    float* Bs     = smem + kAFloats;      // [64][80]  staged router-weight chunk
    float* logits = smem;                 // [128][65] aliases stage region (after GEMM)
    float* gates  = smem + kStageFloats;  // [128][2]

    const int tid    = threadIdx.x;
    const int wave   = tid >> 5;
    const int lane   = tid & 31;
    const int l16    = lane & 15;
    const int h      = lane >> 4;         // lane half selects K pair (A) / K rows (B)
    const int token0 = blockIdx.x * kTokPerWG;

    // 4 accumulator tiles: 16 tokens x (4 x 16) experts, f32
    v8f acc[4];
#pragma unroll
    for (int i = 0; i < 4; ++i)
#pragma unroll
        for (int r = 0; r < 8; ++r) acc[i][r] = 0.0f;

    // ---------------- Phase 1: logits GEMM via V_WMMA_F32_16X16X4_F32 ----------------
    for (int kc = 0; kc < kD; kc += kKC) {
        __syncthreads();
        // Stage A: 128 tokens x 64 features, coalesced b128, 8 per thread
#pragma unroll
        for (int j = 0; j < (kTokPerWG * kKC / 4) / kThreads; ++j) {
            int v = tid + j * kThreads;           // 0..2047
            int t = v >> 4;                       // token 0..127
            int c = (v & 15) << 2;                // col 0..60 step 4
            v4f val = *(const v4f*)(x + (size_t)(token0 + t) * kD + kc + c);
            *(v4f*)(As + t * kAStride + c) = val;
        }
        // Stage B: 64 k-rows x 64 experts, coalesced b128, 4 per thread
#pragma unroll
        for (int j = 0; j < (kKC * kE / 4) / kThreads; ++j) {
            int v = tid + j * kThreads;           // 0..1023
            int k = v >> 4;                       // 0..63
            int c = (v & 15) << 2;                // 0..60 step 4
            v4f val = *(const v4f*)(w + (size_t)(kc + k) * kE + c);
            *(v4f*)(Bs + k * kBStride + c) = val;
        }
        __syncthreads();

        // Each wave: its 16 tokens against all 64 experts
        const float* arow = As + (wave * 16 + l16) * kAStride;
#pragma unroll 4
        for (int kq = 0; kq < kKC; kq += 4) {
            // A frag (16x4 f32): lanes 0-15 -> K={kq,kq+1}; lanes 16-31 -> K={kq+2,kq+3}
            v2f a = *(const v2f*)(arow + kq + 2 * h);
#pragma unroll
            for (int nt = 0; nt < 4; ++nt) {
                // B frag (4x16 f32): vgpr j, half h -> K = kq + 2h + j, N = nt*16 + l16
                v2f b;
                b.x = Bs[(kq + 2 * h)     * kBStride + nt * 16 + l16];
                b.y = Bs[(kq + 2 * h + 1) * kBStride + nt * 16 + l16];
                acc[nt] = __builtin_amdgcn_wmma_f32_16x16x4_f32(
                    false, a, false, b, (short)0, acc[nt], false, false);
            }
        }
    }

    // ---------------- Phase 2: spill logits, softmax + top-2 per token ----------------
    __syncthreads();
#pragma unroll
    for (int nt = 0; nt < 4; ++nt)
#pragma unroll
        for (int r = 0; r < 8; ++r) {
            // C/D layout: vgpr r, lanes 0-15 -> M=r, lanes 16-31 -> M=r+8; N = lane%16
            int tl = wave * 16 + h * 8 + r;
            logits[tl * kLStride + nt * 16 + l16] = acc[nt][r];
        }
    __syncthreads();

    if (tid < kTokPerWG) {
        const float* lrow = logits + tid * kLStride;
        float m1 = -3.4e38f, m2 = -3.4e38f;
        int i1 = 0, i2 = 0;
        for (int e = 0; e < kE; ++e) {
            float v = lrow[e];
            if (v > m1)      { m2 = m1; i2 = i1; m1 = v; i1 = e; }
            else if (v > m2) { m2 = v;  i2 = e; }
        }
        float s = 0.0f;
        for (int e = 0; e < kE; ++e) s += __expf(lrow[e] - m1);
        float inv = 1.0f / s;
        gates[tid * 2 + 0] = inv;                    // exp(m1-m1)/sum
        gates[tid * 2 + 1] = __expf(m2 - m1) * inv;  // second expert prob
        int gt = token0 + tid;
        idx_out[2 * gt + 0] = i1;
        idx_out[2 * gt + 1] = i2;
    }
    __syncthreads();

    // ---------------- Phase 3: gated dispatch (token duplicated x2), NT streaming stores
    for (int t = 0; t < kTokPerWG; ++t) {
        int gt = token0 + t;
        float g1 = gates[2 * t + 0];
        float g2 = gates[2 * t + 1];
        // 256 threads x float4 = 1024 floats: full row, coalesced (L2-resident re-read)
        v4f v = *(const v4f*)(x + (size_t)gt * kD + tid * 4);
        v4f o1 = v * g1;
        v4f o2 = v * g2;
        __builtin_nontemporal_store(o1, (v4f*)(out + (size_t)(2 * gt)     * kD + tid * 4));
        __builtin_nontemporal_store(o2, (v4f*)(out + (size_t)(2 * gt + 1) * kD + tid * 4));
    }
}

extern "C" void kernel_launch(void* const* d_in, const int* in_sizes, int n_in,
                              void* d_out, int out_size, void* d_ws, size_t ws_size,
                              hipStream_t stream) {
    (void)in_sizes; (void)n_in; (void)d_ws; (void)ws_size; (void)out_size;
    const float* x = (const float*)d_in[0];   // [8,4096,1024] f32
    const float* w = (const float*)d_in[1];   // [1024,64] f32
    float* out = (float*)d_out;               // [65536,1024] f32, then 65536 int32 idx
    int* idx_out = (int*)(out + (size_t)kTokens * 2 * kD);
    hipLaunchKernelGGL(moe_router_dispatch,
                       dim3(kTokens / kTokPerWG), dim3(kThreads), 0, stream,
                       x, w, out, idx_out);
}